// BidirectNNF_32134945308758
// MI455X (gfx1250) — compile-verified
//
#include <hip/hip_runtime.h>

// BDS vote for MI455X (gfx1250, wave32).
//   C=64, H=W=512, patch=3, ws=1/(512*512), wr=2/(512*512)
// ws layout: ref_t 64MB | gacc 64MB | wacc 1MB  (all L2-resident: 129MB < 192MB L2)

typedef __attribute__((ext_vector_type(2))) float v2f;
typedef __attribute__((ext_vector_type(8))) float v8f;

constexpr int H = 512, W = 512, C = 64;
constexpr int HWPX = H * W;
constexpr float WGT_S = 1.0f / (512.0f * 512.0f);
constexpr float WGT_R = 2.0f / (512.0f * 512.0f);

// ---- WMMA identity-transpose helper -----------------------------------------
// Feed a 16x16 f32 tile as four 16x4 A-blocks against shifted-identity B blocks:
// D[m][n] = T[m][n], but data moves from A-layout (row per lane) to D-layout
// (row per VGPR pair-half, column per lane) == register-level transpose on the
// matrix pipe. a0/a1 are T[m][4j+2*half + {0,1}] for this lane.
__device__ __forceinline__ v8f wmma_ident_step(v8f acc, float a0, float a1,
                                               int n, int k0) {
  v2f a; a.x = a0; a.y = a1;
  v2f b; b.x = (n == k0) ? 1.0f : 0.0f; b.y = (n == k0 + 1) ? 1.0f : 0.0f;
  return __builtin_amdgcn_wmma_f32_16x16x4_f32(false, a, false, b,
                                               (short)0, acc, false, false);
}

// ---- Kernel 0: ref (C,H,W) -> ref_t (H*W, C) via WMMA transpose -------------
__global__ void __launch_bounds__(256)
k_transpose_ref(const float* __restrict__ ref, float* __restrict__ ref_t) {
  const int wave = (blockIdx.x * blockDim.x + threadIdx.x) >> 5;
  const int lane = threadIdx.x & 31;
  const int m = lane & 15;          // tile row (input), tile col (output)
  const int half = lane >> 4;
  const int c0 = (wave & 3) << 4;   // 64/16 = 4 channel tiles
  const int p0 = (wave >> 2) << 4;  // pixel tile base
  const float* src = ref + (size_t)(c0 + m) * HWPX + p0;  // row m = channel
  v8f acc = {};
#pragma unroll
  for (int j = 0; j < 4; ++j) {
    const int k0 = 4 * j + 2 * half;
    acc = wmma_ident_step(acc, src[k0], src[k0 + 1], m, k0);
  }
  // lane holds pixel p0+m, channels c0 + 8*half + v (v=0..7): 32B contiguous
  float* dst = ref_t + (size_t)(p0 + m) * C + c0 + 8 * half;
  *(float4*)(dst)     = make_float4(acc[0], acc[1], acc[2], acc[3]);
  *(float4*)(dst + 4) = make_float4(acc[4], acc[5], acc[6], acc[7]);
}

// ---- Kernel 1: src->ref pass, inverted to a gather (no atomics) -------------
// One wave per output pixel t; lane covers 2 channels (float2, 256B/wave).
__global__ void __launch_bounds__(256)
k_pass1(const int2* __restrict__ nnf_sr, const float* __restrict__ ref_t,
        float* __restrict__ gacc, float* __restrict__ wacc) {
  const int wave = (blockIdx.x * blockDim.x + threadIdx.x) >> 5;
  const int lane = threadIdx.x & 31;
  const int ty = wave >> 9, tx = wave & 511;
  v2f acc = {0.0f, 0.0f};
  float wsum = 0.0f;
#pragma unroll
  for (int dy = -1; dy <= 1; ++dy) {
#pragma unroll
    for (int dx = -1; dx <= 1; ++dx) {
      const int sy = ty - dy, sx = tx - dx;       // t = s + d  =>  s = t - d
      if ((unsigned)sy < (unsigned)H && (unsigned)sx < (unsigned)W) {
        const int2 nn = nnf_sr[sy * W + sx];      // (ry, rx), lane-uniform
        const int qy = nn.x + dy, qx = nn.y + dx;
        if ((unsigned)qy < (unsigned)H && (unsigned)qx < (unsigned)W) {
          const v2f r =
              *(const v2f*)(ref_t + (size_t)(qy * W + qx) * C + lane * 2);
          acc.x += WGT_S * r.x;
          acc.y += WGT_S * r.y;
          wsum += WGT_S;
        }
      }
    }
  }
  *(v2f*)(gacc + (size_t)wave * C + lane * 2) = acc;  // initializing write
  if (lane == 0) wacc[wave] = wsum;
}

// ---- Kernel 2: ref->src pass, true scatter with f32 atomics (L2-resident) ---
__global__ void __launch_bounds__(256)
k_pass2(const int2* __restrict__ nnf_rs, const float* __restrict__ ref_t,
        float* __restrict__ gacc, float* __restrict__ wacc) {
  const int wave = (blockIdx.x * blockDim.x + threadIdx.x) >> 5;
  const int lane = threadIdx.x & 31;
  const int ry = wave >> 9, rx = wave & 511;
  const int2 nn = nnf_rs[wave];                   // (sy, sx), lane-uniform
#pragma unroll
  for (int dy = -1; dy <= 1; ++dy) {
#pragma unroll
    for (int dx = -1; dx <= 1; ++dx) {
      const int qy = ry + dy, qx = rx + dx;
      const int ty = nn.x + dy, tx = nn.y + dx;
      if ((unsigned)qy < (unsigned)H && (unsigned)qx < (unsigned)W &&
          (unsigned)ty < (unsigned)H && (unsigned)tx < (unsigned)W) {
        const v2f r =
            *(const v2f*)(ref_t + (size_t)(qy * W + qx) * C + lane * 2);
        float* dst = gacc + (size_t)(ty * W + tx) * C + lane * 2;
        unsafeAtomicAdd(dst,     WGT_R * r.x);    // global_atomic_add_f32
        unsafeAtomicAdd(dst + 1, WGT_R * r.y);
        if (lane == 0) unsafeAtomicAdd(wacc + (ty * W + tx), WGT_R);
      }
    }
  }
}

// ---- Kernel 3: normalize + (H*W,C) -> (C,H,W) via WMMA transpose ------------
__global__ void __launch_bounds__(256)
k_norm_out(const float* __restrict__ gacc, const float* __restrict__ wacc,
           float* __restrict__ out) {
  const int wave = (blockIdx.x * blockDim.x + threadIdx.x) >> 5;
  const int lane = threadIdx.x & 31;
  const int m = lane & 15;
  const int half = lane >> 4;
  const int c0 = (wave & 3) << 4;
  const int p0 = (wave >> 2) << 4;
  const float w = wacc[p0 + m];
  const float rw = (w == 0.0f) ? 1.0f : 1.0f / w;
  const float* src = gacc + (size_t)(p0 + m) * C + c0;  // row m = pixel
  v8f acc = {};
#pragma unroll
  for (int j = 0; j < 4; ++j) {
    const int k0 = 4 * j + 2 * half;
    acc = wmma_ident_step(acc, src[k0] * rw, src[k0 + 1] * rw, m, k0);
  }
  // lane holds channel c0+m, pixels p0 + 8*half + v (v=0..7): 32B contiguous
  float* dst = out + (size_t)(c0 + m) * HWPX + p0 + 8 * half;
  *(float4*)(dst)     = make_float4(acc[0], acc[1], acc[2], acc[3]);
  *(float4*)(dst + 4) = make_float4(acc[4], acc[5], acc[6], acc[7]);
}

// -----------------------------------------------------------------------------
extern "C" void kernel_launch(void* const* d_in, const int* in_sizes, int n_in,
                              void* d_out, int out_size, void* d_ws,
                              size_t ws_size, hipStream_t stream) {
  (void)in_sizes; (void)n_in; (void)out_size; (void)ws_size;
  const float* ref    = (const float*)d_in[0];
  const int2* nnf_sr  = (const int2*)d_in[1];
  const int2* nnf_rs  = (const int2*)d_in[2];
  float* out   = (float*)d_out;
  float* ref_t = (float*)d_ws;                     // 64 MB
  float* gacc  = ref_t + (size_t)C * HWPX;         // 64 MB
  float* wacc  = gacc + (size_t)C * HWPX;          // 1 MB

  // 65536 16x16 tiles, 8 waves/block
  k_transpose_ref<<<8192, 256, 0, stream>>>(ref, ref_t);
  // 262144 pixels, 1 wave each
  k_pass1<<<32768, 256, 0, stream>>>(nnf_sr, ref_t, gacc, wacc);
  k_pass2<<<32768, 256, 0, stream>>>(nnf_rs, ref_t, gacc, wacc);
  k_norm_out<<<8192, 256, 0, stream>>>(gacc, wacc, out);
}